// DecoderStage_14113262535155
// MI455X (gfx1250) — compile-verified
//
#include <hip/hip_runtime.h>
#include <math.h>

typedef float v2f __attribute__((ext_vector_type(2)));
typedef float v8f __attribute__((ext_vector_type(8)));

#define BB    2
#define LL    1024      // 32*32 tokens after patch expand
#define DI    768
#define ODIM  384
#define NST   16
#define RR    48
#define KK    4
#define CTOT  80        // R + 2N
#define CIN   768

__device__ __forceinline__ float silu_f(float x) { return x / (1.f + __expf(-x)); }
__device__ __forceinline__ float softplus_f(float x) { return x > 20.f ? x : log1pf(__expf(x)); }

// ---------------------------------------------------------------------------
// Register-tiled batched GEMM via V_WMMA_F32_16X16X4_F32.
//   O[b][m][n] = sum_k A[b][m][k] * W[b%wmod][n][k]   (+ bias / softplus / +=)
// A addressed as A + b*sAb + m*sArow + k*SAK  (SAK is compile-time so the
// common SAK==1 case emits single b64 loads per operand pair).
// W row-major (N x Kd), contiguous, ld = Kd.
// mode: 0 = store (+bias), 1 = softplus(x+bias), 2 = accumulate into O.
// One wave per (16*MT x 16*NT) tile; per K-step: MT+NT b64 loads feed MT*NT
// wmma ops.  grid = (N/(16*NT), M/(16*MT), batch); blockDim = 32 (one wave,
// EXEC all ones as WMMA requires).
// ---------------------------------------------------------------------------
template <int MT, int NT, int SAK>
__global__ void gemm_wmma_tiled(const float* __restrict__ A, long long sAb, int sArow,
                                const float* __restrict__ Wm, long long sWb, int wmod,
                                const float* __restrict__ bias, long long sBb,
                                float* __restrict__ O, long long sOb, int ldo,
                                int Kd, int mode)
{
    const int lane = threadIdx.x & 31;
    const int half = lane >> 4;     // 0: K pair {0,1}, 1: K pair {2,3}
    const int lm   = lane & 15;
    const int n0 = blockIdx.x * 16 * NT;
    const int m0 = blockIdx.y * 16 * MT;
    const int bz = blockIdx.z;

    const float* Ab[MT];
    const float* Wb[NT];
#pragma unroll
    for (int mt = 0; mt < MT; ++mt)
        Ab[mt] = A + (long long)bz * sAb + (long long)(m0 + mt * 16 + lm) * sArow;
#pragma unroll
    for (int nt = 0; nt < NT; ++nt)
        Wb[nt] = Wm + (long long)(bz % wmod) * sWb + (long long)(n0 + nt * 16 + lm) * Kd;

    v8f acc[MT][NT];
#pragma unroll
    for (int mt = 0; mt < MT; ++mt)
#pragma unroll
        for (int nt = 0; nt < NT; ++nt)
            acc[mt][nt] = (v8f){0.f, 0.f, 0.f, 0.f, 0.f, 0.f, 0.f, 0.f};

    for (int k = 0; k < Kd; k += 4) {
        const int ka = k + 2 * half;
        v2f a[MT], b[NT];
#pragma unroll
        for (int mt = 0; mt < MT; ++mt) {
            if (SAK == 1) {
                a[mt] = *reinterpret_cast<const v2f*>(Ab[mt] + ka);   // one b64 load
            } else {
                a[mt].x = Ab[mt][(long long)ka * SAK];
                a[mt].y = Ab[mt][(long long)(ka + 1) * SAK];
            }
        }
#pragma unroll
        for (int nt = 0; nt < NT; ++nt)
            b[nt] = *reinterpret_cast<const v2f*>(Wb[nt] + ka);       // one b64 load
#pragma unroll
        for (int mt = 0; mt < MT; ++mt)
#pragma unroll
            for (int nt = 0; nt < NT; ++nt)
                acc[mt][nt] = __builtin_amdgcn_wmma_f32_16x16x4_f32(
                    false, a[mt], false, b[nt], (short)0, acc[mt][nt], false, false);
    }

    float* Ob = O + (long long)bz * sOb;
    const float* bb = bias ? (bias + (long long)(bz % wmod) * sBb) : nullptr;
#pragma unroll
    for (int nt = 0; nt < NT; ++nt) {
        const int ncol = n0 + nt * 16 + lm;
        const float badd = bb ? bb[ncol] : 0.f;
#pragma unroll
        for (int mt = 0; mt < MT; ++mt) {
#pragma unroll
            for (int v = 0; v < 8; ++v) {
                const int mrow = m0 + mt * 16 + v + 8 * half;  // C/D layout: VGPR v -> M, M+8
                const long long oi = (long long)mrow * ldo + ncol;
                const float x = acc[mt][nt][v];
                if (mode == 0)      Ob[oi] = x + badd;
                else if (mode == 1) Ob[oi] = softplus_f(x + badd);
                else                Ob[oi] += x;
            }
        }
    }
}

// ---------------------------------------------------------------------------
// LayerNorm over last dim C (one block of 256 threads per token).
// ---------------------------------------------------------------------------
__global__ void ln_kernel(const float* __restrict__ in, const float* __restrict__ w,
                          const float* __restrict__ b, float* __restrict__ out, int C)
{
    const int t = blockIdx.x;
    const float* xi = in + (long long)t * C;
    float* yo = out + (long long)t * C;
    __shared__ float s1[256], s2[256];
    float a = 0.f, q = 0.f;
    for (int c = threadIdx.x; c < C; c += 256) { float v = xi[c]; a += v; q += v * v; }
    s1[threadIdx.x] = a; s2[threadIdx.x] = q; __syncthreads();
    for (int o = 128; o > 0; o >>= 1) {
        if ((int)threadIdx.x < o) { s1[threadIdx.x] += s1[threadIdx.x + o];
                                    s2[threadIdx.x] += s2[threadIdx.x + o]; }
        __syncthreads();
    }
    const float mean = s1[0] / (float)C;
    const float var  = s2[0] / (float)C - mean * mean;
    const float rstd = rsqrtf(var + 1e-5f);
    for (int c = threadIdx.x; c < C; c += 256)
        yo[c] = (xi[c] - mean) * rstd * w[c] + b[c];
}

// ---------------------------------------------------------------------------
// Patch-expand rearrange + LN + concat skip -> cat (B*L, 768)
// tmp is the pe GEMM output, (B, 256, 1536) token-major.
// ---------------------------------------------------------------------------
__global__ void pe_rearrange_ln_cat(const float* __restrict__ tmp,
                                    const float* __restrict__ skip,
                                    const float* __restrict__ nw, const float* __restrict__ nb,
                                    float* __restrict__ cat)
{
    const int t = blockIdx.x;            // 0..2047
    const int b = t >> 10, l = t & 1023;
    const int y = l >> 5, x = l & 31;
    const int h = y >> 1, p = y & 1, w = x >> 1, q = x & 1;
    const float* src = tmp + ((long long)(b * 256 + h * 16 + w)) * 1536 + (p * 2 + q) * 384;
    float* dst = cat + (long long)t * 768;
    __shared__ float s1[256], s2[256];
    float a = 0.f, qq = 0.f;
    for (int c = threadIdx.x; c < 384; c += 256) { float v = src[c]; a += v; qq += v * v; }
    s1[threadIdx.x] = a; s2[threadIdx.x] = qq; __syncthreads();
    for (int o = 128; o > 0; o >>= 1) {
        if ((int)threadIdx.x < o) { s1[threadIdx.x] += s1[threadIdx.x + o];
                                    s2[threadIdx.x] += s2[threadIdx.x + o]; }
        __syncthreads();
    }
    const float mean = s1[0] * (1.f / 384.f);
    const float var  = s2[0] * (1.f / 384.f) - mean * mean;
    const float rstd = rsqrtf(var + 1e-5f);
    for (int c = threadIdx.x; c < 384; c += 256)
        dst[c] = (src[c] - mean) * rstd * nw[c] + nb[c];
    for (int c = threadIdx.x; c < 384; c += 256)
        dst[384 + c] = skip[((long long)b * 384 + c) * 1024 + l];
}

// ---------------------------------------------------------------------------
// Depthwise 3x3 conv (SAME) + bias + SiLU.  Input xz token-major (B*L,1536),
// channels 0..767 are xc.  Output xc channel-major (B, DI, L).
// ---------------------------------------------------------------------------
__global__ void conv_silu(const float* __restrict__ xz, const float* __restrict__ cw,
                          const float* __restrict__ cb, float* __restrict__ xc)
{
    const int idx = blockIdx.x * 256 + threadIdx.x;   // B*DI*L
    const int l = idx & 1023;
    const int d = (idx >> 10) % DI;
    const int b = idx / (DI * LL);
    const int h = l >> 5, wp = l & 31;
    float s = cb[d];
#pragma unroll
    for (int ky = 0; ky < 3; ++ky) {
        const int yy = h + ky - 1;
        if (yy < 0 || yy >= 32) continue;
#pragma unroll
        for (int kx = 0; kx < 3; ++kx) {
            const int xx = wp + kx - 1;
            if (xx < 0 || xx >= 32) continue;
            s += cw[d * 9 + ky * 3 + kx] * xz[((long long)(b * LL + yy * 32 + xx)) * 1536 + d];
        }
    }
    xc[((long long)(b * DI + d)) * LL + l] = silu_f(s);
}

// ---------------------------------------------------------------------------
// Build xsT (B,K,L,DI) token-major from xc (B,DI,L) with the 4 scan orders.
// ---------------------------------------------------------------------------
__global__ void gather_xs(const float* __restrict__ xc, float* __restrict__ xsT)
{
    const long long idx = (long long)blockIdx.x * 256 + threadIdx.x;  // B*K*L*DI
    const int d = (int)(idx % DI);
    const int l = (int)((idx / DI) % LL);
    const int g = (int)(idx / ((long long)DI * LL));
    const int b = g >> 2, k = g & 3;
    const int lt = ((l & 31) << 5) | (l >> 5);      // (h,w) <-> (w,h)
    int src;
    if (k == 0)      src = l;
    else if (k == 1) src = lt;
    else if (k == 2) src = 1023 - l;
    else             src = 1023 - lt;
    xsT[idx] = xc[((long long)(b * DI + d)) * LL + src];
}

// ---------------------------------------------------------------------------
// Selective scan: one thread per (g, d); N=16 state in registers.
// B/C rows read as aligned float4 (xdbl row = 320 B, 16-B aligned).
// ---------------------------------------------------------------------------
__global__ void scan_kernel(const float* __restrict__ xsT, const float* __restrict__ dts,
                            const float* __restrict__ xdbl,
                            const float* __restrict__ Alog, const float* __restrict__ Dsw,
                            float* __restrict__ ysb)
{
    const int d = blockIdx.x * 256 + threadIdx.x;  // < DI
    const int g = blockIdx.y;                      // 0..7
    const int k = g & 3;
    float Av[NST];
#pragma unroll
    for (int n = 0; n < NST; ++n)
        Av[n] = -__expf(Alog[((long long)(k * DI + d)) * NST + n]);
    const float Dv = Dsw[k * DI + d];
    float h[NST];
#pragma unroll
    for (int n = 0; n < NST; ++n) h[n] = 0.f;
    const long long tokbase = (long long)g * LL;
    for (int l = 0; l < LL; ++l) {
        const long long tok = tokbase + l;
        const float dt = dts[tok * DI + d];
        const float u  = xsT[tok * DI + d];
        const float4* bc4 = reinterpret_cast<const float4*>(xdbl + tok * CTOT + 48);
        float Bv[NST], Cv[NST];
#pragma unroll
        for (int j = 0; j < 4; ++j) {
            const float4 vb = bc4[j];
            Bv[4 * j] = vb.x; Bv[4 * j + 1] = vb.y; Bv[4 * j + 2] = vb.z; Bv[4 * j + 3] = vb.w;
        }
#pragma unroll
        for (int j = 0; j < 4; ++j) {
            const float4 vc = bc4[4 + j];
            Cv[4 * j] = vc.x; Cv[4 * j + 1] = vc.y; Cv[4 * j + 2] = vc.z; Cv[4 * j + 3] = vc.w;
        }
        const float dtu = dt * u;
        float acc = 0.f;
#pragma unroll
        for (int n = 0; n < NST; ++n) {
            h[n] = __expf(dt * Av[n]) * h[n] + dtu * Bv[n];
            acc += h[n] * Cv[n];
        }
        ysb[tok * DI + d] = acc + Dv * u;
    }
}

// ---------------------------------------------------------------------------
// Combine the 4 directional outputs back to hw order: ycomb (B,L,DI).
// ---------------------------------------------------------------------------
__global__ void combine_y(const float* __restrict__ ysb, float* __restrict__ ycomb)
{
    const long long idx = (long long)blockIdx.x * 256 + threadIdx.x;  // B*L*DI
    const int d = (int)(idx % DI);
    const int l = (int)((idx / DI) % LL);
    const int b = (int)(idx / ((long long)DI * LL));
    const int lt = ((l & 31) << 5) | (l >> 5);
    const float* base = ysb + (long long)b * 4 * LL * DI;
    const float s = base[(long long)l * DI + d]
                  + base[(long long)LL * DI + (long long)lt * DI + d]
                  + base[2ll * LL * DI + (long long)(1023 - l) * DI + d]
                  + base[3ll * LL * DI + (long long)(1023 - lt) * DI + d];
    ycomb[idx] = s;
}

// ---------------------------------------------------------------------------
// out-LN over DI then multiply by silu(z): gated (B*L, DI).
// ---------------------------------------------------------------------------
__global__ void ln_gate(const float* __restrict__ in, const float* __restrict__ w,
                        const float* __restrict__ b, const float* __restrict__ zbase,
                        int zld, float* __restrict__ out)
{
    const int t = blockIdx.x;
    const float* xi = in + (long long)t * DI;
    const float* zi = zbase + (long long)t * zld;
    float* yo = out + (long long)t * DI;
    __shared__ float s1[256], s2[256];
    float a = 0.f, q = 0.f;
    for (int c = threadIdx.x; c < DI; c += 256) { float v = xi[c]; a += v; q += v * v; }
    s1[threadIdx.x] = a; s2[threadIdx.x] = q; __syncthreads();
    for (int o = 128; o > 0; o >>= 1) {
        if ((int)threadIdx.x < o) { s1[threadIdx.x] += s1[threadIdx.x + o];
                                    s2[threadIdx.x] += s2[threadIdx.x + o]; }
        __syncthreads();
    }
    const float mean = s1[0] * (1.f / (float)DI);
    const float var  = s2[0] * (1.f / (float)DI) - mean * mean;
    const float rstd = rsqrtf(var + 1e-5f);
    for (int c = threadIdx.x; c < DI; c += 256)
        yo[c] = ((xi[c] - mean) * rstd * w[c] + b[c]) * silu_f(zi[c]);
}

// ---------------------------------------------------------------------------
// Final (B,L,384) -> (B,384,L) transpose.
// ---------------------------------------------------------------------------
__global__ void final_transpose(const float* __restrict__ xh, float* __restrict__ out)
{
    const int idx = blockIdx.x * 256 + threadIdx.x;   // B*384*1024
    const int l = idx & 1023;
    const int c = (idx >> 10) % ODIM;
    const int b = idx / (ODIM * LL);
    out[idx] = xh[((long long)(b * LL + l)) * ODIM + c];
}

extern "C" void kernel_launch(void* const* d_in, const int* in_sizes, int n_in,
                              void* d_out, int out_size, void* d_ws, size_t ws_size,
                              hipStream_t stream)
{
    const float* x      = (const float*)d_in[0];   // (2,768,16,16)
    const float* skip   = (const float*)d_in[1];   // (2,384,32,32)
    const float* peW    = (const float*)d_in[2];   // (1536,768)
    const float* peNw   = (const float*)d_in[3];
    const float* peNb   = (const float*)d_in[4];
    const float* lpW    = (const float*)d_in[5];   // (384,768)
    const float* lpB    = (const float*)d_in[6];
    const float* blkW   = (const float*)d_in[7];   // (2,384)
    const float* blkB   = (const float*)d_in[8];
    const float* inW    = (const float*)d_in[9];   // (2,1536,384)
    const float* convW  = (const float*)d_in[10];  // (2,768,3,3)
    const float* convB  = (const float*)d_in[11];  // (2,768)
    const float* xpW    = (const float*)d_in[12];  // (2,4,80,768)
    const float* dtW    = (const float*)d_in[13];  // (2,4,768,48)
    const float* dtB    = (const float*)d_in[14];  // (2,4,768)
    const float* Alog   = (const float*)d_in[15];  // (2,4,768,16)
    const float* Dsw    = (const float*)d_in[16];  // (2,4,768)
    const float* onW    = (const float*)d_in[17];  // (2,768)
    const float* onB    = (const float*)d_in[18];
    const float* opW    = (const float*)d_in[19];  // (2,384,768)

    float* ws    = (float*)d_ws;
    float* xh    = ws;                 // 786432   (B*L*384)  residual stream
    float* hnorm = ws + 786432;        // 786432
    float* xz    = ws + 1572864;       // 3145728  (B*L*1536) [also pe tmp]
    float* xc    = ws + 4718592;       // 1572864  (B,DI,L)   [also cat buffer]
    float* xdbl  = ws + 6291456;       // 655360   (B*K,L,80)
    float* dts   = ws + 6946816;       // 6291456  (B*K,L,DI)
    float* ysb   = ws + 13238272;      // 6291456  (B*K,L,DI)
    float* xsT   = ws + 19529728;      // 6291456  (B*K,L,DI)
    float* ycomb = xsT;                // reuse after scan
    float* gated = xsT + 1572864;      // reuse after scan

    // 0a. Patch-expand GEMM: x (channel-major, k-stride 256) @ peW^T -> (B,256,1536)
    gemm_wmma_tiled<2, 4, 256><<<dim3(1536 / 64, 256 / 32, BB), 32, 0, stream>>>(
        x, 768ll * 256, 1, peW, 0, 1, nullptr, 0,
        xz, 256ll * 1536, 1536, 768, 0);
    // 0b. rearrange + LN + concat skip -> cat in xc (B*L,768)
    pe_rearrange_ln_cat<<<BB * LL, 256, 0, stream>>>(xz, skip, peNw, peNb, xc);
    // 0c. lp: cat @ lpW^T + lpB -> xh (B*L,384)
    gemm_wmma_tiled<2, 4, 1><<<dim3(ODIM / 64, BB * LL / 32, 1), 32, 0, stream>>>(
        xc, 0, 768, lpW, 0, 1, lpB, 0, xh, 0, ODIM, 768, 0);

    for (int i = 0; i < 2; ++i) {
        // LN
        ln_kernel<<<BB * LL, 256, 0, stream>>>(xh, blkW + i * ODIM, blkB + i * ODIM,
                                               hnorm, ODIM);
        // in_proj: (B*L,384) @ (1536,384)^T -> xz (B*L,1536)
        gemm_wmma_tiled<2, 4, 1><<<dim3(1536 / 64, BB * LL / 32, 1), 32, 0, stream>>>(
            hnorm, 0, ODIM, inW + (long long)i * 1536 * ODIM, 0, 1, nullptr, 0,
            xz, 0, 1536, ODIM, 0);
        // depthwise conv + SiLU -> xc (B,DI,L)
        conv_silu<<<BB * DI * LL / 256, 256, 0, stream>>>(
            xz, convW + (long long)i * DI * 9, convB + (long long)i * DI, xc);
        // 4-direction gather -> xsT (B*K,L,DI)
        gather_xs<<<BB * KK * LL * DI / 256, 256, 0, stream>>>(xc, xsT);
        // x_dbl: per (b,k): (L,768) @ (80,768)^T -> (B*K,L,80)   (N=80 -> 2x1 tile)
        gemm_wmma_tiled<2, 1, 1><<<dim3(CTOT / 16, LL / 32, BB * KK), 32, 0, stream>>>(
            xsT, (long long)LL * DI, DI,
            xpW + (long long)i * KK * CTOT * DI, (long long)CTOT * DI, KK,
            nullptr, 0, xdbl, (long long)LL * CTOT, CTOT, DI, 0);
        // dts: softplus(dtr @ dtW^T + dtB) -> (B*K,L,DI)
        gemm_wmma_tiled<2, 4, 1><<<dim3(DI / 64, LL / 32, BB * KK), 32, 0, stream>>>(
            xdbl, (long long)LL * CTOT, CTOT,
            dtW + (long long)i * KK * DI * RR, (long long)DI * RR, KK,
            dtB + (long long)i * KK * DI, DI,
            dts, (long long)LL * DI, DI, RR, 1);
        // selective scan -> ysb (B*K,L,DI)
        scan_kernel<<<dim3(DI / 256, BB * KK), 256, 0, stream>>>(
            xsT, dts, xdbl,
            Alog + (long long)i * KK * DI * NST, Dsw + (long long)i * KK * DI, ysb);
        // combine 4 directions -> ycomb (B*L,DI)
        combine_y<<<BB * LL * DI / 256, 256, 0, stream>>>(ysb, ycomb);
        // out-LN * silu(z) -> gated
        ln_gate<<<BB * LL, 256, 0, stream>>>(ycomb, onW + (long long)i * DI,
                                             onB + (long long)i * DI,
                                             xz + DI, 1536, gated);
        // out_proj accumulate into residual xh
        gemm_wmma_tiled<2, 4, 1><<<dim3(ODIM / 64, BB * LL / 32, 1), 32, 0, stream>>>(
            gated, 0, DI, opW + (long long)i * ODIM * DI, 0, 1, nullptr, 0,
            xh, 0, ODIM, DI, 2);
    }

    // final transpose (B,L,384) -> (B,384,32,32)
    final_transpose<<<BB * ODIM * LL / 256, 256, 0, stream>>>(xh, (float*)d_out);
}